// OperationScorer_16655883174705
// MI455X (gfx1250) — compile-verified
//
#include <hip/hip_runtime.h>

typedef __attribute__((ext_vector_type(2))) float v2f;
typedef __attribute__((ext_vector_type(8))) float v8f;
typedef __attribute__((ext_vector_type(4))) int v4i;

#define B_ 8
#define T_ 512
#define S_ 512
#define M_ 512
#define H_ 128
// 1/sqrt(512)
#define SCALE_INV 0.04419417382415922f

#if __has_builtin(__builtin_amdgcn_global_load_async_to_lds_b128)
#define USE_ASYNC_LDS 1
#else
#define USE_ASYNC_LDS 0
#endif

__device__ __forceinline__ float htanh(float x) {
#if __has_builtin(__builtin_amdgcn_tanhf)
  return __builtin_amdgcn_tanhf(x);
#elif __has_builtin(__builtin_amdgcn_tanh_f32)
  return __builtin_amdgcn_tanh_f32(x);
#else
  float r;
  asm("v_tanh_f32 %0, %1" : "=v"(r) : "v"(x));
  return r;
#endif
}

#if USE_ASYNC_LDS
typedef __attribute__((address_space(1))) v4i gv4i;  // global int4
typedef __attribute__((address_space(3))) v4i lv4i;  // LDS int4

__device__ __forceinline__ void async_b128_to_lds(const float* g, float* l) {
  __builtin_amdgcn_global_load_async_to_lds_b128((gv4i*)g, (lv4i*)l, 0, 0);
}
__device__ __forceinline__ void wait_async_le2() {
#if __has_builtin(__builtin_amdgcn_s_wait_asynccnt)
  __builtin_amdgcn_s_wait_asynccnt(2);
#else
  asm volatile("s_wait_asynccnt 0x2" ::: "memory");
#endif
}
__device__ __forceinline__ void wait_async_0() {
#if __has_builtin(__builtin_amdgcn_s_wait_asynccnt)
  __builtin_amdgcn_s_wait_asynccnt(0);
#else
  asm volatile("s_wait_asynccnt 0x0" ::: "memory");
#endif
}
#endif

// C[R x 128] = A[R x 512] * W[512 x 128], exact f32 via V_WMMA_F32_16X16X4_F32.
// One wave per 16x16 output tile; 8 waves (256 threads) per block.
__global__ __launch_bounds__(256) void proj_wmma(const float* __restrict__ A,
                                                 const float* __restrict__ W,
                                                 float* __restrict__ X, int R) {
  const int lane = threadIdx.x & 31;
  const int wave = threadIdx.x >> 5;
  const int tile = blockIdx.x * 8 + wave;
  const int mt = tile >> 3;  // H/16 == 8 n-tiles
  const int nt = tile & 7;
  if ((mt << 4) >= R) return;  // wave-uniform: EXEC stays all-ones for WMMA

  // A 16x4 f32 layout: lanes 0-15 hold M=lane, K={0,1}; lanes 16-31 K={2,3}
  const int mrow = lane & 15;
  const int koff = (lane >> 4) << 1;
  const int ncol = lane & 15;

  const float* Arow = A + (size_t)(mt * 16 + mrow) * M_;
  const float* Wcol = W + nt * 16 + ncol;

  v8f c = {};
#pragma unroll 8
  for (int k0 = 0; k0 < M_; k0 += 4) {
    v2f a = *(const v2f*)(Arow + k0 + koff);  // global_load_b64
    v2f b;
    b.x = Wcol[(size_t)(k0 + koff) * H_];
    b.y = Wcol[(size_t)(k0 + koff + 1) * H_];
    c = __builtin_amdgcn_wmma_f32_16x16x4_f32(false, a, false, b, (short)0, c,
                                              false, false);
  }

  // D layout: VGPR r, lane l -> m = r + 8*(l>=16), n = l&15
  float* Xp = X + (size_t)(mt * 16) * H_ + nt * 16 + ncol;
  const int mhi = (lane >> 4) << 3;
#pragma unroll
  for (int r = 0; r < 8; ++r) Xp[(size_t)(r + mhi) * H_] = c[r];
}

// score[b,t,s] = (1/sqrt(512)) * sum_h v[h] * tanh(x1[b,t,h] + x2[b,s,h])
// One block per (b, 16-row t-tile); stream 16-row s-chunks through LDS,
// double-buffered via GLOBAL_LOAD_ASYNC_TO_LDS_B128 (ASYNCcnt) when available.
__global__ __launch_bounds__(256) void score_kernel(
    const float* __restrict__ X1, const float* __restrict__ X2,
    const float* __restrict__ v, float* __restrict__ out) {
  __shared__ __attribute__((aligned(16))) float sX1[16][132];
  __shared__ __attribute__((aligned(16))) float sX2[2][16][132];
  __shared__ __attribute__((aligned(16))) float sv[H_];

  const int tid = threadIdx.x;
  const int b = blockIdx.x >> 5;  // T/16 == 32 t-tiles
  const int tt = blockIdx.x & 31;

  // cooperative load pattern: 16 rows x 8 floats/thread (two b128 per thread)
  const int row = tid >> 4;
  const int h8 = (tid & 15) * 8;

  const float* x1g = X1 + ((size_t)b * T_ + tt * 16 + row) * H_ + h8;
  *(float4*)&sX1[row][h8] = *(const float4*)(x1g);
  *(float4*)&sX1[row][h8 + 4] = *(const float4*)(x1g + 4);
  if (tid < H_ / 4) ((float4*)sv)[tid] = ((const float4*)v)[tid];

  const int s_local = tid & 15;  // fast index -> coalesced stores over s
  const int t_local = tid >> 4;

  // this thread's global source within each 16-row s-chunk
  const float* x2g = X2 + ((size_t)b * S_ + row) * H_ + h8;

#if USE_ASYNC_LDS
  // prologue: chunk 0 -> buffer 0
  async_b128_to_lds(x2g, &sX2[0][row][h8]);
  async_b128_to_lds(x2g + 4, &sX2[0][row][h8 + 4]);
#endif

  const int NP = S_ / 16;  // 32 passes
  for (int p = 0; p < NP; ++p) {
    const int s0 = p * 16;
#if USE_ASYNC_LDS
    __syncthreads();  // everyone done reading buf[(p+1)&1] (pass p-1 compute)
    if (p + 1 < NP) {
      const float* g = x2g + (size_t)(s0 + 16) * H_;
      async_b128_to_lds(g, &sX2[(p + 1) & 1][row][h8]);
      async_b128_to_lds(g + 4, &sX2[(p + 1) & 1][row][h8 + 4]);
      wait_async_le2();  // chunk p's 2 loads done (in-order completion)
    } else {
      wait_async_0();
    }
    __syncthreads();  // all waves' chunk-p LDS writes visible
    const float4* r2 = (const float4*)&sX2[p & 1][s_local][0];
#else
    __syncthreads();
    const float* g = x2g + (size_t)s0 * H_;
    *(float4*)&sX2[0][row][h8] = *(const float4*)(g);
    *(float4*)&sX2[0][row][h8 + 4] = *(const float4*)(g + 4);
    __syncthreads();
    const float4* r2 = (const float4*)&sX2[0][s_local][0];
#endif

    float acc = 0.f;
    const float4* r1 = (const float4*)&sX1[t_local][0];
    const float4* rv = (const float4*)sv;
#pragma unroll
    for (int q = 0; q < H_ / 4; ++q) {
      float4 a = r1[q];
      float4 bb = r2[q];
      float4 w = rv[q];
      acc = fmaf(htanh(a.x + bb.x), w.x, acc);
      acc = fmaf(htanh(a.y + bb.y), w.y, acc);
      acc = fmaf(htanh(a.z + bb.z), w.z, acc);
      acc = fmaf(htanh(a.w + bb.w), w.w, acc);
    }
    out[((size_t)b * T_ + tt * 16 + t_local) * S_ + s0 + s_local] =
        acc * SCALE_INV;
  }
}

extern "C" void kernel_launch(void* const* d_in, const int* in_sizes, int n_in,
                              void* d_out, int out_size, void* d_ws,
                              size_t ws_size, hipStream_t stream) {
  const float* query = (const float*)d_in[0];  // (B,T,M)
  const float* keys = (const float*)d_in[1];   // (B,S,N)
  const float* W1 = (const float*)d_in[2];     // (N,H)
  const float* W2 = (const float*)d_in[3];     // (M,H)
  const float* v = (const float*)d_in[4];      // (H)
  float* out = (float*)d_out;

  float* x1 = (float*)d_ws;               // B*T*H floats (2 MB)
  float* x2 = x1 + (size_t)B_ * T_ * H_;  // B*S*H floats (2 MB)

  const int R = B_ * T_;  // 4096 rows
  dim3 blk(256);
  dim3 grd((R / 16) * (H_ / 16) / 8);  // 2048 tiles / 8 waves per block
  proj_wmma<<<grd, blk, 0, stream>>>(query, W2, x1, R);
  proj_wmma<<<grd, blk, 0, stream>>>(keys, W1, x2, R);
  score_kernel<<<dim3(B_ * (T_ / 16)), blk, 0, stream>>>(x1, x2, v, out);
}